// NeuralTuringMachine_59356448031059
// MI455X (gfx1250) — compile-verified
//
#include <hip/hip_runtime.h>
#include <cstddef>
#include <cstdint>

// NTM single step, fp32 WMMA (v_wmma_f32_16x16x4_f32) implementation for gfx1250.
// B=256, IN=512, H=1024, OUT=512, S=512, Wd=256, SH=3, P_READ=262, P_WRITE=774.

typedef __attribute__((ext_vector_type(2))) float v2f;
typedef __attribute__((ext_vector_type(8))) float v8f;

#define Bsz 256
#define INsz 512
#define Hsz 1024
#define Ssz 512
#define Wdsz 256
#define PREAD 262
#define PWRITE 774

__device__ __forceinline__ float sigmoidf(float x) { return 1.0f / (1.0f + __expf(-x)); }
__device__ __forceinline__ float softplusf(float x) {
  if (x > 20.0f) return x;
  if (x < -20.0f) return __expf(x);
  return log1pf(__expf(x));
}

// ---------------------------------------------------------------------------
// Generic fp32 WMMA GEMM:  C[M x N] = A[M x K] @ op(B) + bias
//   TRANSB=1: op(B)[k][n] = Bm[n*ldb + k]   (weights stored row-major [N,K])
//   TRANSB=0: op(B)[k][n] = Bm[k*ldb + n]   (plain [K,N] matrix)
// One wave -> one 16x16 tile via V_WMMA_F32_16X16X4_F32, K stepped by 4.
// Block = 256 threads = 8 waves = 16(M) x 128(N) per block. M must be %16.
// ---------------------------------------------------------------------------
template <int TRANSB>
__global__ void wmma_gemm_f32(const float* __restrict__ A, int lda,
                              const float* __restrict__ Bm, int ldb,
                              const float* __restrict__ bias,
                              float* __restrict__ C, int ldc,
                              int Ntot, int K) {
  const int wave = threadIdx.x >> 5;
  const int lane = threadIdx.x & 31;
  const int half = lane >> 4;   // 0: K pair {0,1} / M rows 0..7 ; 1: K pair {2,3} / M rows 8..15
  const int idx  = lane & 15;   // M row for A-frag, N col for B-frag and C/D

  const int mb = blockIdx.y * 16;
  const int n0 = (blockIdx.x * 8 + wave) * 16;
  const int n  = n0 + idx;
  const bool nval = (n < Ntot);

  const float* Arow = A + (size_t)(mb + idx) * lda;     // per-lane A row (16x4 frag layout)
  const float* Brow = TRANSB ? (Bm + (size_t)(nval ? n : 0) * ldb) : Bm;

  __builtin_prefetch(Arow, 0, 3);
  if (nval) __builtin_prefetch(Brow, 0, 3);

  v8f acc = {};
  for (int k = 0; k < K; k += 4) {
    const int kk = k + half * 2;
    // A fragment: lanes 0-15 hold M=0..15 / K={k,k+1}; lanes 16-31 K={k+2,k+3}
    v2f a = *(const v2f*)(Arow + kk);
    // B fragment: lanes 0-15 hold N cols / K={k,k+1}; lanes 16-31 K={k+2,k+3}
    v2f b = {0.0f, 0.0f};
    if (nval) {
      if (TRANSB) {
        b = *(const v2f*)(Brow + kk);
      } else {
        b.x = Bm[(size_t)kk * ldb + n];
        b.y = Bm[(size_t)(kk + 1) * ldb + n];
      }
    }
    acc = __builtin_amdgcn_wmma_f32_16x16x4_f32(false, a, false, b,
                                                (short)0, acc, false, false);
  }

  if (nval) {
    const float bv = bias ? bias[n] : 0.0f;
#pragma unroll
    for (int r = 0; r < 8; ++r) {
      const int m = r + half * 8;  // C/D layout: VGPR r holds M=r (lanes 0-15) / M=r+8 (lanes 16-31)
      C[(size_t)(mb + m) * ldc + n] = acc[r] + bv;
    }
  }
}

// ---------------------------------------------------------------------------
// Fused LSTM gate GEMM + activation, async-LDS staged A panel.
// h_prev = c_prev = 0 so:
//   only gates i (rows j), g (rows j+2H), o (rows j+3H) of W_ih are needed,
//   only the first IN columns of W_ih (rv part of input is zero),
//   c = sigmoid(i)*tanh(g); h = sigmoid(o)*tanh(c).
// The 16(M) x 512(K) x-panel is shared by all 8 waves of the block, so it is
// DMA'd into LDS once per 64-wide K chunk with GLOBAL_LOAD_ASYNC_TO_LDS_B128
// (ASYNCcnt), double-buffered so the next chunk's DMA overlaps this chunk's
// 48 WMMAs. Row pitch 68 floats -> the 16-lane fragment reads hit 16 banks.
// Block 256 thr = 8 waves covering 16(M) x 128(H cols). Grid (8, 16).
// ---------------------------------------------------------------------------
#define KCHUNK 64
#define APITCH 68
#define NCHUNK (INsz / KCHUNK)

__device__ __forceinline__ void issue_async_a_tile(const float* __restrict__ xg,
                                                   float* lds_tile, int t) {
  // thread t of 256 moves 16 bytes: row = t>>4 (0..15), seg = t&15 (0..15)
  const int row = t >> 4;
  const int seg = t & 15;
  const float* g = xg + (size_t)row * INsz + seg * 4;           // 16B aligned
  const unsigned lds = (unsigned)(uintptr_t)(lds_tile + row * APITCH + seg * 4);
  asm volatile("global_load_async_to_lds_b128 %0, %1, off"
               :: "v"(lds), "v"(g)
               : "memory");
}

__global__ void lstm_gates_kernel(const float* __restrict__ x,
                                  const float* __restrict__ W_ih,
                                  const float* __restrict__ b_ih,
                                  const float* __restrict__ b_hh,
                                  float* __restrict__ hcat) {
  __shared__ float Atile[2][16 * APITCH];

  const int t    = threadIdx.x;
  const int wave = t >> 5;
  const int lane = t & 31;
  const int half = lane >> 4;
  const int idx  = lane & 15;

  const int mb = blockIdx.y * 16;
  const int j  = (blockIdx.x * 8 + wave) * 16 + idx;  // hidden column, always < 1024

  const int ldw = INsz + Wdsz;  // 768
  const float* xg = x + (size_t)mb * INsz;
  const float* Wi = W_ih + (size_t)(j)           * ldw;
  const float* Wg = W_ih + (size_t)(j + 2 * Hsz) * ldw;
  const float* Wo = W_ih + (size_t)(j + 3 * Hsz) * ldw;

  __builtin_prefetch(Wi, 0, 3);
  __builtin_prefetch(Wg, 0, 3);
  __builtin_prefetch(Wo, 0, 3);

  // prologue: DMA chunk 0
  issue_async_a_tile(xg, &Atile[0][0], t);

  v8f ai = {}, ag = {}, ao = {};
  for (int c = 0; c < NCHUNK; ++c) {
    const int buf = c & 1;
    const int kc  = c * KCHUNK;
    if (c + 1 < NCHUNK) {
      // DMA next chunk into the other buffer, then wait for *this* chunk only
      issue_async_a_tile(xg + (kc + KCHUNK), &Atile[buf ^ 1][0], t);
      asm volatile("s_wait_asynccnt 1" ::: "memory");
    } else {
      asm volatile("s_wait_asynccnt 0" ::: "memory");
    }
    __syncthreads();  // whole 16x64 tile (all waves' DMAs) now visible in LDS

    const float* As = &Atile[buf][(size_t)idx * APITCH];
#pragma unroll
    for (int kl = 0; kl < KCHUNK; kl += 4) {
      const int kk = kl + half * 2;
      v2f a  = *(const v2f*)(As + kk);           // ds_load_b64, bank-conflict-free
      v2f bi = *(const v2f*)(Wi + kc + kk);
      v2f bg = *(const v2f*)(Wg + kc + kk);
      v2f bo = *(const v2f*)(Wo + kc + kk);
      ai = __builtin_amdgcn_wmma_f32_16x16x4_f32(false, a, false, bi, (short)0, ai, false, false);
      ag = __builtin_amdgcn_wmma_f32_16x16x4_f32(false, a, false, bg, (short)0, ag, false, false);
      ao = __builtin_amdgcn_wmma_f32_16x16x4_f32(false, a, false, bo, (short)0, ao, false, false);
    }
    __syncthreads();  // all waves done reading buf before it is re-DMA'd
  }

  const float bi_v = b_ih[j]           + b_hh[j];
  const float bg_v = b_ih[j + 2 * Hsz] + b_hh[j + 2 * Hsz];
  const float bo_v = b_ih[j + 3 * Hsz] + b_hh[j + 3 * Hsz];

#pragma unroll
  for (int r = 0; r < 8; ++r) {
    const int m = r + half * 8;
    const float gi = ai[r] + bi_v;
    const float gg = ag[r] + bg_v;
    const float go = ao[r] + bo_v;
    const float cc = sigmoidf(gi) * tanhf(gg);
    const float h  = sigmoidf(go) * tanhf(cc);
    hcat[(size_t)(mb + m) * (Hsz + Wdsz) + j] = h;
  }
}

// mn[s] = max(||mem_bias[s,:]||, 1e-8)
__global__ void mem_norm_kernel(const float* __restrict__ mem, float* __restrict__ mn) {
  const int s = blockIdx.x * blockDim.x + threadIdx.x;
  if (s < Ssz) {
    const float* row = mem + (size_t)s * Wdsz;
    float acc = 0.0f;
    for (int w = 0; w < Wdsz; ++w) { const float v = row[w]; acc += v * v; }
    mn[s] = fmaxf(sqrtf(acc), 1e-8f);
  }
}

// ---------------------------------------------------------------------------
// NTM addressing: one block (512 thr) per batch row; thread s owns slot s.
//   cw = softmax_s( key.mem[s] / (||key|| ||mem[s]||) * beta )
//   gw = g*cw + (1-g)*onehot0 ; sw = circular 3-tap shift ; w = sw^gamma / sum
// num[b,s] = key.mem[s] was precomputed on the matrix cores.
// ---------------------------------------------------------------------------
__global__ void address_kernel(const float* __restrict__ p, int ldp,
                               const float* __restrict__ num,
                               const float* __restrict__ mn,
                               float* __restrict__ wout) {
  __shared__ float red[Ssz];
  __shared__ float gws[Ssz];
  __shared__ float sc[7];
  const int b = blockIdx.x;
  const int s = threadIdx.x;
  const float* pb = p + (size_t)b * ldp;

  // ||key||
  float v = (s < Wdsz) ? pb[s] : 0.0f;
  red[s] = v * v;
  __syncthreads();
  for (int off = Ssz / 2; off > 0; off >>= 1) {
    if (s < off) red[s] += red[s + off];
    __syncthreads();
  }
  if (s == 0) {
    sc[0] = fmaxf(sqrtf(red[0]), 1e-8f);          // kn
    sc[1] = softplusf(pb[Wdsz]);                  // beta
    sc[2] = sigmoidf(pb[Wdsz + 1]);               // gate
    const float a0 = pb[Wdsz + 2], a1 = pb[Wdsz + 3], a2 = pb[Wdsz + 4];
    const float mx = fmaxf(a0, fmaxf(a1, a2));
    const float e0 = __expf(a0 - mx), e1 = __expf(a1 - mx), e2 = __expf(a2 - mx);
    const float es = e0 + e1 + e2;
    sc[3] = e0 / es; sc[4] = e1 / es; sc[5] = e2 / es;  // shift weights
    sc[6] = 1.0f + softplusf(pb[Wdsz + 5]);       // gamma
  }
  __syncthreads();
  const float kn = sc[0], beta = sc[1], g = sc[2];
  const float sh0 = sc[3], sh1 = sc[4], sh2 = sc[5], gamma = sc[6];

  const float sim = num[(size_t)b * Ssz + s] / (kn * mn[s]) * beta;

  // softmax over S
  red[s] = sim;
  __syncthreads();
  for (int off = Ssz / 2; off > 0; off >>= 1) {
    if (s < off) red[s] = fmaxf(red[s], red[s + off]);
    __syncthreads();
  }
  const float mx = red[0];
  __syncthreads();
  const float e = __expf(sim - mx);
  red[s] = e;
  __syncthreads();
  for (int off = Ssz / 2; off > 0; off >>= 1) {
    if (s < off) red[s] += red[s + off];
    __syncthreads();
  }
  const float cw = e / red[0];
  __syncthreads();

  // interpolate with one-hot prev_w, circular shift, sharpen
  const float gw = g * cw + (1.0f - g) * ((s == 0) ? 1.0f : 0.0f);
  gws[s] = gw;
  __syncthreads();
  const float sw = sh0 * gws[(s + Ssz - 1) & (Ssz - 1)] + sh1 * gw + sh2 * gws[(s + 1) & (Ssz - 1)];
  const float sp = powf(sw, gamma);
  red[s] = sp;
  __syncthreads();
  for (int off = Ssz / 2; off > 0; off >>= 1) {
    if (s < off) red[s] += red[s + off];
    __syncthreads();
  }
  wout[(size_t)b * Ssz + s] = sp / (red[0] + 1e-6f);
}

// rw[b,s] = read_w*write_w ; t3[b] = sum_s rw
__global__ void rw_prep_kernel(const float* __restrict__ rw_in,
                               const float* __restrict__ ww_in,
                               float* __restrict__ rwprod,
                               float* __restrict__ t3) {
  __shared__ float red[Ssz];
  const int b = blockIdx.x;
  const int s = threadIdx.x;
  const float pr = rw_in[(size_t)b * Ssz + s] * ww_in[(size_t)b * Ssz + s];
  rwprod[(size_t)b * Ssz + s] = pr;
  red[s] = pr;
  __syncthreads();
  for (int off = Ssz / 2; off > 0; off >>= 1) {
    if (s < off) red[s] += red[s + off];
    __syncthreads();
  }
  if (s == 0) t3[b] = red[0];
}

// rv[b,w] = t1 - erase*t2 + add*t3 ; written into hcat[:, 1024:1280]
__global__ void rv_finalize_kernel(const float* __restrict__ wp,
                                   const float* __restrict__ t1,
                                   const float* __restrict__ t2,
                                   const float* __restrict__ t3,
                                   float* __restrict__ hcat) {
  const int b = blockIdx.x;
  const int w = threadIdx.x;
  const float er = sigmoidf(wp[(size_t)b * PWRITE + PREAD + w]);
  const float ad = tanhf(wp[(size_t)b * PWRITE + PREAD + Wdsz + w]);
  const float rv = t1[(size_t)b * Wdsz + w] - er * t2[(size_t)b * Wdsz + w] + ad * t3[b];
  hcat[(size_t)b * (Hsz + Wdsz) + Hsz + w] = rv;
}

// ---------------------------------------------------------------------------
extern "C" void kernel_launch(void* const* d_in, const int* in_sizes, int n_in,
                              void* d_out, int out_size, void* d_ws, size_t ws_size,
                              hipStream_t stream) {
  (void)in_sizes; (void)n_in; (void)out_size; (void)ws_size;
  const float* x        = (const float*)d_in[0];
  const float* W_ih     = (const float*)d_in[1];
  const float* b_ih     = (const float*)d_in[2];
  // d_in[3] = W_hh: unused (h_prev == 0)
  const float* b_hh     = (const float*)d_in[4];
  const float* W_read   = (const float*)d_in[5];
  const float* b_read   = (const float*)d_in[6];
  const float* W_write  = (const float*)d_in[7];
  const float* b_write  = (const float*)d_in[8];
  const float* W_out    = (const float*)d_in[9];
  const float* b_out    = (const float*)d_in[10];
  const float* mem_bias = (const float*)d_in[11];
  float* out = (float*)d_out;

  // workspace layout (floats)
  float* ws      = (float*)d_ws;
  float* hcat    = ws;                            // 256 x 1280 [h | rv]
  float* rp      = hcat   + Bsz * (Hsz + Wdsz);   // 256 x 262
  float* wp      = rp     + Bsz * PREAD;          // 256 x 774
  float* num_r   = wp     + Bsz * PWRITE;         // 256 x 512
  float* num_w   = num_r  + Bsz * Ssz;            // 256 x 512
  float* read_w  = num_w  + Bsz * Ssz;            // 256 x 512
  float* write_w = read_w + Bsz * Ssz;            // 256 x 512
  float* rwprod  = write_w+ Bsz * Ssz;            // 256 x 512
  float* t1      = rwprod + Bsz * Ssz;            // 256 x 256
  float* t2      = t1     + Bsz * Wdsz;           // 256 x 256
  float* t3      = t2     + Bsz * Wdsz;           // 256
  float* mn      = t3     + Bsz;                  // 512

  const dim3 blk(256);

  // memory row norms (batch-invariant)
  mem_norm_kernel<<<dim3(2), blk, 0, stream>>>(mem_bias, mn);

  // LSTM: h into hcat[:, :1024]
  lstm_gates_kernel<<<dim3(Hsz / 128, Bsz / 16), blk, 0, stream>>>(x, W_ih, b_ih, b_hh, hcat);

  // rp = h @ W_read.T + b_read ; wp = h @ W_write.T + b_write  (K = 1024 cols of hcat)
  wmma_gemm_f32<1><<<dim3((PREAD  + 127) / 128, Bsz / 16), blk, 0, stream>>>(
      hcat, Hsz + Wdsz, W_read,  Hsz, b_read,  rp, PREAD,  PREAD,  Hsz);
  wmma_gemm_f32<1><<<dim3((PWRITE + 127) / 128, Bsz / 16), blk, 0, stream>>>(
      hcat, Hsz + Wdsz, W_write, Hsz, b_write, wp, PWRITE, PWRITE, Hsz);

  // cosine numerators: num = key @ mem_bias.T  (key = first Wd cols of rp/wp)
  wmma_gemm_f32<1><<<dim3(Ssz / 128, Bsz / 16), blk, 0, stream>>>(
      rp, PREAD,  mem_bias, Wdsz, nullptr, num_r, Ssz, Ssz, Wdsz);
  wmma_gemm_f32<1><<<dim3(Ssz / 128, Bsz / 16), blk, 0, stream>>>(
      wp, PWRITE, mem_bias, Wdsz, nullptr, num_w, Ssz, Ssz, Wdsz);

  // addressing softmax pipelines
  address_kernel<<<dim3(Bsz), dim3(Ssz), 0, stream>>>(rp, PREAD,  num_r, mn, read_w);
  address_kernel<<<dim3(Bsz), dim3(Ssz), 0, stream>>>(wp, PWRITE, num_w, mn, write_w);

  // rw product + t3
  rw_prep_kernel<<<dim3(Bsz), dim3(Ssz), 0, stream>>>(read_w, write_w, rwprod, t3);

  // t1 = read_w @ mem_bias ; t2 = rw @ mem_bias   (plain [K,N] B-matrix)
  wmma_gemm_f32<0><<<dim3(Wdsz / 128, Bsz / 16), blk, 0, stream>>>(
      read_w, Ssz, mem_bias, Wdsz, nullptr, t1, Wdsz, Wdsz, Ssz);
  wmma_gemm_f32<0><<<dim3(Wdsz / 128, Bsz / 16), blk, 0, stream>>>(
      rwprod, Ssz, mem_bias, Wdsz, nullptr, t2, Wdsz, Wdsz, Ssz);

  // rv into hcat[:, 1024:1280]
  rv_finalize_kernel<<<dim3(Bsz), dim3(Wdsz), 0, stream>>>(wp, t1, t2, t3, hcat);

  // out = [h, rv] @ W_out.T + b_out
  wmma_gemm_f32<1><<<dim3(512 / 128, Bsz / 16), blk, 0, stream>>>(
      hcat, Hsz + Wdsz, W_out, Hsz + Wdsz, b_out, out, 512, 512, Hsz + Wdsz);
}